// WEIGHTSPEmbedderAR_21062519620290
// MI455X (gfx1250) — compile-verified
//
#include <hip/hip_runtime.h>

// B=512 graphs, N=512 nodes/graph, C=64 channels, L=4 layers, 8192 edges/graph.
// One workgroup (256 threads = 8 waves) per graph; fully fused 4-layer pipeline.
// LDS: x-buffer 128KB + agg-buffer (pitch 66) 132KB + W 16KB + small scratch
//  = ~285KB of the 320KB WGP LDS. prev-layer h lives in d_ws (B*N*C*4 = 64MB).

#define BG   512
#define NN   512
#define CC   64
#define LL   4
#define EDG  (NN * 16)   // 8192 edges per graph
#define AGP  66          // agg buffer row pitch (breaks stride-64 bank conflicts)
#define EPSV 1e-5f
#define SLOPEV 0.01f

typedef __attribute__((ext_vector_type(2))) float v2f;
typedef __attribute__((ext_vector_type(8))) float v8f;

__device__ __forceinline__ float lrelu(float x) { return x > 0.f ? x : SLOPEV * x; }

__global__ __launch_bounds__(256)
void gnn_fused_kernel(const float* __restrict__ node_feats,
                      const float* __restrict__ wglob,
                      const int*   __restrict__ esrc,
                      const int*   __restrict__ edst,
                      const float* __restrict__ AR,
                      const float* __restrict__ convW,
                      const float* __restrict__ gamma_,
                      const float* __restrict__ beta_,
                      const float* __restrict__ alpha_,
                      float* __restrict__ out,
                      float* __restrict__ prevbuf)
{
    extern __shared__ float smem[];
    float* xbuf   = smem;                   // NN*CC   (32768)
    float* aggbuf = xbuf   + NN * CC;       // NN*AGP  (33792)
    float* wmat   = aggbuf + NN * AGP;      // CC*CC   (4096)
    float* oinv   = wmat   + CC * CC;       // NN
    float* iinv   = oinv   + NN;            // NN
    float* wnode  = iinv   + NN;            // NN
    float* sum1   = wnode  + NN;            // 256
    float* sum2   = sum1   + 256;           // 256
    float* red    = sum2   + 256;           // 256
    float* mulc   = red    + 256;           // CC
    float* addc   = mulc   + CC;            // CC

    const int t     = threadIdx.x;
    const int g     = blockIdx.x;
    const int nbase = g * NN;
    const long base64 = (long)nbase * CC;
    const int* eS = esrc + (long)g * EDG;
    const int* eD = edst + (long)g * EDG;

    // ---- degrees (LDS int atomics, aliased on aggbuf) + load node feats ----
    int* ocnt = (int*)aggbuf;
    int* icnt = ocnt + NN;
    for (int i = t; i < 2 * NN; i += 256) ocnt[i] = 0;
    for (int i = t; i < NN * CC; i += 256) xbuf[i] = node_feats[base64 + i];
    __syncthreads();
    for (int e = t; e < EDG; e += 256) {
        atomicAdd(&ocnt[eS[e] - nbase], 1);
        atomicAdd(&icnt[eD[e] - nbase], 1);
    }
    __syncthreads();
    for (int n = t; n < NN; n += 256) {
        int oc = ocnt[n]; if (oc < 1) oc = 1;
        int ic = icnt[n]; if (ic < 1) ic = 1;
        oinv[n] = rsqrtf((float)oc);   // norm='both', clamp deg >= 1
        iinv[n] = rsqrtf((float)ic);
    }

    const int lane = t & 31;
    const int wv   = t >> 5;        // wave 0..7
    const int half = lane >> 4;     // 0/1
    const int lr   = lane & 15;
    const int cch  = t & 63;        // channel of this thread
    const int ng   = t >> 6;        // node/edge slot group 0..3

    for (int l = 0; l < LL; ++l) {
        __syncthreads();
        // per-layer constants into LDS; zero agg; pre-scale x by out_inv
        for (int i = t; i < CC * CC; i += 256) wmat[i] = convW[l * CC * CC + i];
        for (int n = t; n < NN; n += 256) wnode[n] = wglob[nbase + n] * AR[l * NN + n];
        for (int i = t; i < NN * AGP; i += 256) aggbuf[i] = 0.f;
        for (int i = t; i < NN * CC; i += 256) xbuf[i] *= oinv[i >> 6];
        __syncthreads();

        // ---- edge aggregation: agg[dst] += (x*out_inv)[src]  (LDS fp32 atomics) ----
        for (int e0 = 0; e0 < EDG; e0 += 4) {
            int e = e0 + ng;                 // wave-uniform edge index
            int s = eS[e] - nbase;
            int d = eD[e] - nbase;
            atomicAdd(&aggbuf[d * AGP + cch], xbuf[s * CC + cch]);
        }
        __syncthreads();

        // ---- Y = (agg * in_inv) @ W via V_WMMA_F32_16X16X4_F32, in-place ----
        for (int tile = 0; tile < 4; ++tile) {
            const int m0   = wv * 64 + tile * 16;
            const int arow = m0 + lr;
            const float ai = iinv[arow];
            v8f acc0 = {}; v8f acc1 = {}; v8f acc2 = {}; v8f acc3 = {};
            #pragma unroll
            for (int k = 0; k < CC; k += 4) {
                const int ak = k + 2 * half;
                v2f a;
                a.x = aggbuf[arow * AGP + ak]     * ai;
                a.y = aggbuf[arow * AGP + ak + 1] * ai;
                v2f b0, b1, b2, b3;
                b0.x = wmat[ak * CC +  0 + lr]; b0.y = wmat[(ak + 1) * CC +  0 + lr];
                b1.x = wmat[ak * CC + 16 + lr]; b1.y = wmat[(ak + 1) * CC + 16 + lr];
                b2.x = wmat[ak * CC + 32 + lr]; b2.y = wmat[(ak + 1) * CC + 32 + lr];
                b3.x = wmat[ak * CC + 48 + lr]; b3.y = wmat[(ak + 1) * CC + 48 + lr];
                acc0 = __builtin_amdgcn_wmma_f32_16x16x4_f32(false, a, false, b0, (short)0, acc0, false, false);
                acc1 = __builtin_amdgcn_wmma_f32_16x16x4_f32(false, a, false, b1, (short)0, acc1, false, false);
                acc2 = __builtin_amdgcn_wmma_f32_16x16x4_f32(false, a, false, b2, (short)0, acc2, false, false);
                acc3 = __builtin_amdgcn_wmma_f32_16x16x4_f32(false, a, false, b3, (short)0, acc3, false, false);
            }
            const int wrow = m0 + half * 8;   // C/D layout: vgpr r -> M = r (+8 for hi half)
            #pragma unroll
            for (int r = 0; r < 8; ++r) {
                aggbuf[(wrow + r) * AGP +  0 + lr] = acc0[r];
                aggbuf[(wrow + r) * AGP + 16 + lr] = acc1[r];
                aggbuf[(wrow + r) * AGP + 32 + lr] = acc2[r];
                aggbuf[(wrow + r) * AGP + 48 + lr] = acc3[r];
            }
        }
        __syncthreads();

        // ---- GraphNorm stats: per-channel sum(y), sum(y^2) over 512 nodes ----
        float s1 = 0.f, s2 = 0.f;
        for (int i = 0; i < NN / 4; ++i) {
            int n = ng + i * 4;
            float y = aggbuf[n * AGP + cch];
            s1 += y; s2 += y * y;
        }
        sum1[ng * CC + cch] = s1;
        sum2[ng * CC + cch] = s2;
        __syncthreads();
        if (t < CC) {
            float a1 = sum1[t] + sum1[CC + t] + sum1[2 * CC + t] + sum1[3 * CC + t];
            float a2 = sum2[t] + sum2[CC + t] + sum2[2 * CC + t] + sum2[3 * CC + t];
            float mean = a1 * (1.f / NN);
            float ey2  = a2 * (1.f / NN);
            float al   = alpha_[l * CC + t];
            float var  = ey2 - 2.f * al * mean * mean + al * al * mean * mean;
            float rinv = rsqrtf(var + EPSV);
            float gm = gamma_[l * CC + t], bt = beta_[l * CC + t];
            mulc[t] = gm * rinv;
            addc[t] = bt - gm * rinv * al * mean;
        }
        __syncthreads();

        // ---- normalize+lrelu -> h ; residual u = h + prev ; readout partial ----
        const float mc = mulc[cch], ac = addc[cch];
        const float* prevsrc = (l == 0) ? (node_feats + base64) : (prevbuf + base64);
        float r = 0.f;
        for (int i = 0; i < NN / 4; ++i) {
            int n = ng + i * 4;
            float y  = aggbuf[n * AGP + cch];
            float h  = lrelu(mc * y + ac);
            float pv = prevsrc[(long)n * CC + cch];
            float u  = h + pv;
            xbuf[n * CC + cch] = u;                    // conv input for next layer
            prevbuf[base64 + (long)n * CC + cch] = h;  // prev for next residual
            r += wnode[n] * ((l == 0) ? h : u);        // layer-0 readout is pre-residual
        }
        red[ng * CC + cch] = r;
        __syncthreads();
        if (t < CC) {
            float rr = (red[t] + red[CC + t] + red[2 * CC + t] + red[3 * CC + t]) * (1.f / NN);
            out[(long)g * (CC * LL) + l * CC + t] = lrelu(rr);  // final lrelu on concat
        }
    }
}

extern "C" void kernel_launch(void* const* d_in, const int* in_sizes, int n_in,
                              void* d_out, int out_size, void* d_ws, size_t ws_size,
                              hipStream_t stream) {
    const float* node_feats = (const float*)d_in[0];
    const float* weights    = (const float*)d_in[1];
    const int*   esrc       = (const int*)d_in[2];
    const int*   edst       = (const int*)d_in[3];
    const float* AR         = (const float*)d_in[4];
    const float* convW      = (const float*)d_in[5];
    const float* gamma_     = (const float*)d_in[6];
    const float* beta_      = (const float*)d_in[7];
    const float* alpha_     = (const float*)d_in[8];
    float* out     = (float*)d_out;
    float* prevbuf = (float*)d_ws;   // needs B*N*C*4 = 64 MB scratch

    const size_t smem = (size_t)(NN * CC + NN * AGP + CC * CC + 3 * NN + 3 * 256 + 2 * CC)
                        * sizeof(float);  // ~285 KB of the 320 KB WGP LDS
    hipFuncSetAttribute(reinterpret_cast<const void*>(gnn_fused_kernel),
                        hipFuncAttributeMaxDynamicSharedMemorySize, (int)smem);
    gnn_fused_kernel<<<dim3(BG), dim3(256), smem, stream>>>(
        node_feats, weights, esrc, edst, AR, convW, gamma_, beta_, alpha_,
        out, prevbuf);
}